// MolMPNN_3547642987145
// MI455X (gfx1250) — compile-verified
//
#include <hip/hip_runtime.h>

// MolMPNN layer for MI455X (gfx1250, wave32).
//   h_pre = h @ W_pre + b_pre                       (WMMA f32 GEMM)
//   agg   = segment_sum(h_pre[src], dst)            (wave/edge, hw f32 atomics)
//   t     = relu(agg @ W1 + b1)                     (WMMA f32 GEMM)
//   out   = LN(h_pre + relu(t @ W2 + b2))           (WMMA GEMM fused epilogue)

#define D 128
#define LDS_STRIDE 132   // pad: 132%64==4 -> conflict-free b64 LDS reads; 132*4%16==0 -> b128-alignable rows

typedef float v2f __attribute__((ext_vector_type(2)));
typedef float v8f __attribute__((ext_vector_type(8)));
typedef int   v4i __attribute__((ext_vector_type(4)));

#if defined(__has_builtin)
#if __has_builtin(__builtin_amdgcn_global_load_async_to_lds_b128) && \
    __has_builtin(__builtin_amdgcn_s_wait_asynccnt)
#define USE_ASYNC_LDS 1
#endif
#endif

// ---------------------------------------------------------------- GEMM core
// Block = 256 threads = 8 waves; computes rows [row0,row0+16) x 128 cols.
// Wave w owns the 16x16 tile at columns [16w,16w+16).
__device__ __forceinline__ void stage_tiles(const float* __restrict__ in,
                                            const float* __restrict__ W,
                                            float* alds, float* wlds,
                                            int row0, int n_rows, int tid) {
    // W (128x128 row-major [k][n]) -> wlds[n*LDS_STRIDE + k] (transposed), coalesced reads
    #pragma unroll 4
    for (int j = 0; j < 64; ++j) {
        int flat = tid + j * 256;
        int k = flat >> 7, n = flat & 127;
        wlds[n * LDS_STRIDE + k] = W[flat];
    }
    // A tile: 16 rows x 128 cols.  16B chunks: fi = chunk index, 32 chunks per row.
    #pragma unroll
    for (int i = 0; i < 2; ++i) {
        int fi = tid + i * 256;          // 0..511
        int r = fi >> 5, c4 = fi & 31;
        int row = row0 + r;
        int srow = (row < n_rows) ? row : (n_rows - 1);
        const float* gp = in + (size_t)srow * D + c4 * 4;
        float* lp = alds + r * LDS_STRIDE + c4 * 4;   // 16B aligned (stride 132)
#ifdef USE_ASYNC_LDS
        __builtin_amdgcn_global_load_async_to_lds_b128(
            (__attribute__((address_space(1))) v4i*)(uintptr_t)gp,
            (__attribute__((address_space(3))) v4i*)(uintptr_t)lp, 0, 0);
#else
        const float4 v = *(const float4*)gp;
        lp[0] = v.x; lp[1] = v.y; lp[2] = v.z; lp[3] = v.w;
#endif
    }
#ifdef USE_ASYNC_LDS
    __builtin_amdgcn_s_wait_asynccnt(0);
#endif
}

__device__ __forceinline__ v8f wmma_mainloop(const float* alds, const float* wlds,
                                             int lane, int wave) {
    const int m    = lane & 15;
    const int koff = (lane < 16) ? 0 : 2;
    const int n    = wave * 16 + m;
    v8f c = {};
    #pragma unroll 4
    for (int kb = 0; kb < 32; ++kb) {
        const int k = kb * 4 + koff;
        v2f a = *(const v2f*)(alds + m * LDS_STRIDE + k);
        v2f b = *(const v2f*)(wlds + n * LDS_STRIDE + k);
        c = __builtin_amdgcn_wmma_f32_16x16x4_f32(false, a, false, b,
                                                  (short)0, c, false, false);
    }
    return c;
}

// ------------------------------------------------ plain GEMM (+bias, opt relu)
__global__ __launch_bounds__(256) void gemm_kernel(const float* __restrict__ in,
                                                   const float* __restrict__ W,
                                                   const float* __restrict__ bias,
                                                   float* __restrict__ out,
                                                   int n_rows, int do_relu) {
    __shared__ float alds[16 * LDS_STRIDE];
    __shared__ float wlds[D * LDS_STRIDE];
    const int tid  = threadIdx.x;
    const int wave = tid >> 5, lane = tid & 31;
    const int row0 = blockIdx.x * 16;

    stage_tiles(in, W, alds, wlds, row0, n_rows, tid);
    __syncthreads();

    v8f c = wmma_mainloop(alds, wlds, lane, wave);

    const int n     = wave * 16 + (lane & 15);
    const int rbase = row0 + ((lane < 16) ? 0 : 8);
    const float bn  = bias[n];
    float* op = out + (size_t)rbase * D + n;
    if (row0 + 16 <= n_rows) {                 // fast path: no per-row guards
        #pragma unroll
        for (int r = 0; r < 8; ++r) {
            float v = c[r] + bn;
            if (do_relu) v = fmaxf(v, 0.f);
            op[(size_t)r * D] = v;
        }
    } else {
        #pragma unroll
        for (int r = 0; r < 8; ++r) {
            if (rbase + r < n_rows) {
                float v = c[r] + bn;
                if (do_relu) v = fmaxf(v, 0.f);
                op[(size_t)r * D] = v;
            }
        }
    }
}

// --------------------- final GEMM fused: bias + relu + residual + LayerNorm
__global__ __launch_bounds__(256) void gemm_res_ln_kernel(const float* __restrict__ in,
                                                          const float* __restrict__ W,
                                                          const float* __restrict__ bias,
                                                          const float* __restrict__ resid,
                                                          const float* __restrict__ gamma,
                                                          const float* __restrict__ beta,
                                                          float* __restrict__ out,
                                                          int n_rows) {
    __shared__ float alds[16 * LDS_STRIDE];
    __shared__ float wlds[D * LDS_STRIDE];
    __shared__ float otile[16 * D];
    __shared__ float stats[32];      // {mu, rsig} per row
    const int tid  = threadIdx.x;
    const int wave = tid >> 5, lane = tid & 31;
    const int row0 = blockIdx.x * 16;

    stage_tiles(in, W, alds, wlds, row0, n_rows, tid);
    __syncthreads();

    v8f c = wmma_mainloop(alds, wlds, lane, wave);

    const int n     = wave * 16 + (lane & 15);
    const int lrow0 = (lane < 16) ? 0 : 8;
    const float bn  = bias[n];
    const float* rp = resid + (size_t)(row0 + lrow0) * D + n;
    #pragma unroll
    for (int r = 0; r < 8; ++r) {
        int srow = row0 + lrow0 + r;
        float res = (srow < n_rows) ? rp[(size_t)r * D] : 0.f;
        otile[(lrow0 + r) * D + n] = fmaxf(c[r] + bn, 0.f) + res;
    }
    __syncthreads();

    // Row stats: 16 lanes per row, shuffle-reduce within 16-lane groups.
    {
        int row = tid >> 4;
        int t16 = tid & 15;
        float s = 0.f, ss = 0.f;
        #pragma unroll
        for (int i = 0; i < 8; ++i) {
            float v = otile[row * D + t16 * 8 + i];
            s += v; ss += v * v;
        }
        #pragma unroll
        for (int msk = 1; msk < 16; msk <<= 1) {
            s  += __shfl_xor(s,  msk, 16);
            ss += __shfl_xor(ss, msk, 16);
        }
        if (t16 == 0) {
            float mu  = s * (1.f / 128.f);
            float var = ss * (1.f / 128.f) - mu * mu;
            stats[row * 2]     = mu;
            stats[row * 2 + 1] = rsqrtf(var + 1e-5f);
        }
    }
    __syncthreads();

    if (row0 + 16 <= n_rows) {
        #pragma unroll
        for (int i = 0; i < 8; ++i) {         // fully coalesced final store
            int idx = tid + i * 256;
            int row = idx >> 7, col = idx & 127;
            float mu = stats[row * 2], rs = stats[row * 2 + 1];
            out[(size_t)(row0 + row) * D + col] =
                (otile[idx] - mu) * rs * gamma[col] + beta[col];
        }
    } else {
        #pragma unroll
        for (int i = 0; i < 8; ++i) {
            int idx = tid + i * 256;
            int row = idx >> 7, col = idx & 127;
            if (row0 + row < n_rows) {
                float mu = stats[row * 2], rs = stats[row * 2 + 1];
                out[(size_t)(row0 + row) * D + col] =
                    (otile[idx] - mu) * rs * gamma[col] + beta[col];
            }
        }
    }
}

// ---------------------------------------------------------------- utilities
__global__ void zero_kernel(float4* __restrict__ p, long n4) {
    long i = blockIdx.x * (long)blockDim.x + threadIdx.x;
    long stride = (long)gridDim.x * blockDim.x;
    const float4 z = {0.f, 0.f, 0.f, 0.f};
    for (; i < n4; i += stride) p[i] = z;
}

// One wave per edge: scalarized indices, float4 source row read (512B/wave),
// 4 hardware global_atomic_add_f32 per lane into L2-resident agg.
__global__ __launch_bounds__(256) void scatter_kernel(const float* __restrict__ hpre,
                                                      const int* __restrict__ src,
                                                      const int* __restrict__ dst,
                                                      float* __restrict__ agg,
                                                      int n_edges) {
    const int wid  = (blockIdx.x << 3) + (threadIdx.x >> 5);   // edge id (wave-uniform)
    const int lane = threadIdx.x & 31;
    if (wid >= n_edges) return;
    const int s = __builtin_amdgcn_readfirstlane(src[wid]);
    const int d = __builtin_amdgcn_readfirstlane(dst[wid]);
    const float4 v = *(const float4*)(hpre + (size_t)s * D + lane * 4);
    float* ap = agg + (size_t)d * D + lane * 4;
    unsafeAtomicAdd(ap + 0, v.x);
    unsafeAtomicAdd(ap + 1, v.y);
    unsafeAtomicAdd(ap + 2, v.z);
    unsafeAtomicAdd(ap + 3, v.w);
}

// ---------------------------------------------------------------- launcher
extern "C" void kernel_launch(void* const* d_in, const int* in_sizes, int n_in,
                              void* d_out, int out_size, void* d_ws, size_t ws_size,
                              hipStream_t stream) {
    const float* h     = (const float*)d_in[0];
    const int*   src   = (const int*)d_in[1];
    const int*   dst   = (const int*)d_in[2];
    const float* Wpre  = (const float*)d_in[3];
    const float* bpre  = (const float*)d_in[4];
    const float* W1    = (const float*)d_in[5];
    const float* b1    = (const float*)d_in[6];
    const float* W2    = (const float*)d_in[7];
    const float* b2    = (const float*)d_in[8];
    const float* gamma = (const float*)d_in[9];
    const float* beta  = (const float*)d_in[10];

    const int n_nodes = in_sizes[0] / D;
    const int n_edges = in_sizes[1];

    float* hpre = (float*)d_ws;                       // [n_nodes, D]
    float* agg  = hpre + (size_t)n_nodes * D;         // [n_nodes, D]
    float* tbuf = agg  + (size_t)n_nodes * D;         // [n_nodes, D]
    float* out  = (float*)d_out;

    const int nblk = (n_nodes + 15) / 16;

    gemm_kernel<<<nblk, 256, 0, stream>>>(h, Wpre, bpre, hpre, n_nodes, 0);
    zero_kernel<<<2048, 256, 0, stream>>>((float4*)agg, (long)n_nodes * D / 4);
    const int sblk = (n_edges + 7) / 8;               // 8 waves (edges) per block
    scatter_kernel<<<sblk, 256, 0, stream>>>(hpre, src, dst, agg, n_edges);
    gemm_kernel<<<nblk, 256, 0, stream>>>(agg, W1, b1, tbuf, n_nodes, 1);
    gemm_res_ln_kernel<<<nblk, 256, 0, stream>>>(tbuf, W2, b2, hpre, gamma, beta, out, n_nodes);
}